// GPConv1d_72447508349461
// MI455X (gfx1250) — compile-verified
//
#include <hip/hip_runtime.h>

typedef __attribute__((ext_vector_type(2))) float v2f;
typedef __attribute__((ext_vector_type(8))) float v8f;
typedef __attribute__((ext_vector_type(4))) unsigned int v4u;
typedef __attribute__((ext_vector_type(8))) int v8i;
typedef __attribute__((ext_vector_type(4))) int v4i;

#define L_DIM   4096
#define CIN     512     // C * BD
#define COUT    512     // O * Kb
#define KTAPS   5
#define PADL    2

#define OC_TILE 64
#define L_TILE  128
#define IC_CHUNK 16
#define NCHUNK  (CIN / IC_CHUNK)
#define XS_COLS (L_TILE + 2 * PADL)   // 132 columns staged
#define XS_P    18                     // 16 ic + 2 pad (b64 align, bank spread)
#define KS_W    20                     // 16 ic + 4 pad == TDM pad_interval=16dw/pad_amount=4dw

// LDS byte offset of a __shared__ object (generic -> AS(3) -> int)
#define LDS_OFF(p) \
  ((unsigned)(unsigned long long)(__attribute__((address_space(3))) const void*)(p))

// ---------------------------------------------------------------------------
// Fold Cayley table into weights:
//   kern[v][oc][ic] = sum_i G[i, j, kb] * W[v, c, o, i],  oc=o*8+kb, ic=c*8+j
// ---------------------------------------------------------------------------
__global__ void gp_prep_kern(const float* __restrict__ W, const float* __restrict__ G,
                             float* __restrict__ kern) {
  int idx = blockIdx.x * blockDim.x + threadIdx.x;
  if (idx >= KTAPS * COUT * CIN) return;
  int ic = idx % CIN;
  int oc = (idx / CIN) % COUT;
  int v  = idx / (CIN * COUT);
  int j  = ic & 7, c = ic >> 3;
  int kb = oc & 7, o = oc >> 3;
  float s = 0.f;
#pragma unroll
  for (int i = 0; i < 8; ++i)
    s += G[(i * 8 + j) * 8 + kb] * W[((v * 64 + c) * 64 + o) * 8 + i];
  kern[idx] = s;
}

// ---------------------------------------------------------------------------
// TDM: DMA one kern slab [5 taps][64 oc][16 ic] (f32) into LDS.
// 3D tile: X=16 (contiguous), Y=64 rows @ stride CIN, Z=5 slabs @ stride COUT*CIN.
// LDS padding: +4 DWORDs every 16 DWORDs -> rows land on KS_W=20-float stride.
// D# packing per CDNA5 ISA 8.3-8.6. workgroup_mask=0 (no cluster dispatch).
// This toolchain exposes the 6-arg builtin: (v4u, v8i, v4i, v4i, v8i, cpol).
// ---------------------------------------------------------------------------
__device__ __forceinline__ void tdm_load_kern_slab(const float* gsrc, unsigned lds_off) {
  unsigned long long ga = (unsigned long long)(__UINTPTR_TYPE__)gsrc;
  v4u g0;
  g0.x = 1u;                                             // count=1, user descriptor
  g0.y = lds_off;                                        // lds_addr (bytes)
  g0.z = (unsigned)(ga & 0xFFFFFFFFu);                   // global_addr[31:0]
  g0.w = (unsigned)((ga >> 32) & 0x01FFFFFFu) | (2u << 30);  // [56:32] | type=2
  v8i g1;
  g1[0] = (2 << 16)        // data_size: 4B
        | (1 << 20)        // pad_enable
        | (3 << 22)        // pad_interval code 3 -> 16 DWORDs
        | (3 << 25);       // pad_amount  code 3 -> 4 DWORDs
  g1[1] = (int)((unsigned)CIN << 16);                        // tensor_dim0[15:0]
  g1[2] = (int)(((unsigned)CIN >> 16) | ((unsigned)COUT << 16));   // td0 hi | td1 lo
  g1[3] = (int)(((unsigned)COUT >> 16) | (16u << 16));       // td1 hi | tile_dim0=16
  g1[4] = (int)(64u | ((unsigned)KTAPS << 16));              // tile_dim1=64, tile_dim2=5
  g1[5] = CIN;                                               // dim0_stride[31:0]
  g1[6] = (int)((((unsigned)(COUT * CIN)) & 0xFFFFu) << 16); // s0 hi16(=0) | s1[15:0]
  g1[7] = (int)(((unsigned)(COUT * CIN)) >> 16);             // s1[47:16]
  v4i g2 = {KTAPS, 0, 0, 0};                                 // tensor_dim2=5
  v4i g3 = {0, 0, 0, 0};
  v8i g4 = {0, 0, 0, 0, 0, 0, 0, 0};                         // unused trailing group
  __builtin_amdgcn_tensor_load_to_lds(g0, g1, g2, g3, g4, 0);
}

// ---------------------------------------------------------------------------
// Implicit-GEMM conv via V_WMMA_F32_16X16X4_F32.
//   out[b, oc, l] = bias[oc] + sum_{v, ic} kern[v][oc][ic] * x[b, ic, l + v - 2]
// Block: 256 threads (8 wave32). Tile: 64 oc x 128 l. Wave: 16 oc x 64 l.
// Kern slab: TDM double-buffered (wave 0 issues, TENSORcnt-pipelined one ahead).
// x slab: transposed in LDS so each B fragment is one ds_load_b64.
// ---------------------------------------------------------------------------
__global__ __launch_bounds__(256)
void gp_conv_wmma(const float* __restrict__ x, const float* __restrict__ kern,
                  const float* __restrict__ bias, float* __restrict__ out) {
  __shared__ float xS[XS_COLS][XS_P];              // [l-col][ic]        9.5 KB
  __shared__ float kS[2][KTAPS][OC_TILE][KS_W];    // double-buffered   51.2 KB

  const int tid  = threadIdx.x;
  const int lane = tid & 31;
  const int wv   = tid >> 5;        // 0..7
  const int wr   = wv & 3;          // oc subtile: rows wr*16 .. wr*16+15
  const int wc   = wv >> 2;         // l half: cols wc*64 .. wc*64+63
  const int l0   = blockIdx.x * L_TILE;
  const int oc0  = blockIdx.y * OC_TILE;
  const int b    = blockIdx.z;

  // WMMA f32 16x16x4 per-lane fragment coords (ISA 7.12.2):
  //   A: lane holds A[m][k0], A[m][k0+1];  B: lane holds B[k0][n], B[k0+1][n]
  //   C/D: VGPR i = C[(lane>>4)*8 + i][lane&15]
  const int m  = lane & 15;
  const int kk = lane >> 4;
  const int k0 = kk << 1;

  v8f acc[4];
#pragma unroll
  for (int t = 0; t < 4; ++t) acc[t] = (v8f)0.f;

  const float* xb = x + (size_t)b * CIN * L_DIM;
  const float* ktile = kern + (size_t)oc0 * CIN;   // v=0 slab, row oc0

  // Prologue: DMA chunk 0's weights while we set up
  if (tid < 32) tdm_load_kern_slab(ktile, LDS_OFF(&kS[0]));

  for (int ch = 0; ch < NCHUNK; ++ch) {
    const int icc = ch * IC_CHUNK;
    const int p   = ch & 1;
    __syncthreads();   // prior compute done: xS and kS[1-p] are writable

    // ---- Stage x slab (transposed): xS[col][ic], col <-> global l0-2+col ----
    for (int idx = tid; idx < IC_CHUNK * XS_COLS; idx += 256) {
      int r    = idx / XS_COLS;
      int cpos = idx % XS_COLS;
      int gl   = l0 - PADL + cpos;
      float val = 0.f;
      if (gl >= 0 && gl < L_DIM) val = xb[(size_t)(icc + r) * L_DIM + gl];
      xS[cpos][r] = val;
    }

    // ---- Wave 0: issue next chunk's TDM, then drain this chunk's ----
    if (tid < 32) {
      if (ch + 1 < NCHUNK) {
        tdm_load_kern_slab(ktile + (icc + IC_CHUNK), LDS_OFF(&kS[1 - p]));
        __builtin_amdgcn_s_wait_tensorcnt(1);   // oldest (chunk ch) complete
      } else {
        __builtin_amdgcn_s_wait_tensorcnt(0);
      }
    } else if (tid >= 64 && tid < 128 && ch + 1 < NCHUNK) {
      // Waves 2-3: cache-prefetch next x chunk
      int r = (tid - 64) >> 2, seg = tid & 3;
      int gl = l0 - PADL + seg * 34;
      if (gl >= 0 && gl < L_DIM)
        __builtin_prefetch(&xb[(size_t)(icc + IC_CHUNK + r) * L_DIM + gl], 0, 0);
    }
    __syncthreads();   // publish xS and kS[p]

#pragma unroll
    for (int v = 0; v < KTAPS; ++v) {
#pragma unroll
      for (int rc = 0; rc < IC_CHUNK / 4; ++rc) {
        // A fragment: one ds_load_b64 (row stride 20 floats, 8B aligned)
        v2f a = *(const v2f*)&kS[p][v][wr * 16 + m][rc * 4 + k0];
#pragma unroll
        for (int t = 0; t < 4; ++t) {
          // B fragment: one ds_load_b64 (adjacent K in transposed slab)
          v2f bf = *(const v2f*)&xS[wc * 64 + t * 16 + m + v][rc * 4 + k0];
          acc[t] = __builtin_amdgcn_wmma_f32_16x16x4_f32(
              false, a, false, bf, (short)0, acc[t], false, false);
        }
      }
    }
  }

  // ---- Epilogue: bias + coalesced stores ----
  const int mb = kk * 8;
  float bv[8];
#pragma unroll
  for (int i = 0; i < 8; ++i) bv[i] = bias[oc0 + wr * 16 + mb + i];
#pragma unroll
  for (int t = 0; t < 4; ++t) {
    const int col = l0 + wc * 64 + t * 16 + m;
#pragma unroll
    for (int i = 0; i < 8; ++i) {
      const int row = oc0 + wr * 16 + mb + i;
      out[((size_t)b * COUT + row) * L_DIM + col] = acc[t][i] + bv[i];
    }
  }
}

extern "C" void kernel_launch(void* const* d_in, const int* in_sizes, int n_in,
                              void* d_out, int out_size, void* d_ws, size_t ws_size,
                              hipStream_t stream) {
  (void)in_sizes; (void)n_in; (void)out_size; (void)ws_size;
  const float* x = (const float*)d_in[0];
  const float* W = (const float*)d_in[1];
  const float* b = (const float*)d_in[2];
  const float* G = (const float*)d_in[3];
  float* kern = (float*)d_ws;                       // 5*512*512*4 = 5.24 MB scratch

  int total = KTAPS * COUT * CIN;
  gp_prep_kern<<<(total + 255) / 256, 256, 0, stream>>>(W, G, kern);

  dim3 grid(L_DIM / L_TILE, COUT / OC_TILE, 8);
  gp_conv_wmma<<<grid, 256, 0, stream>>>(x, kern, b, (float*)d_out);
}